// AllLoss_41446434406714
// MI455X (gfx1250) — compile-verified
//
#include <hip/hip_runtime.h>

#define NN   8
#define AA   16368
#define KK   200
#define BB   20
#define PP   4
#define HWD  128
#define PIX  (HWD * HWD)          // 16384
#define NTIL (PIX / 16)           // 1024 pixel tiles of 16
#define MTIL ((KK + 15) / 16)     // 13 anchor tiles of 16
#define EPSF 1e-7f
#define BCE_CAP 16.118095651f     // -log(1e-7): clamp(p,eps,1-eps) equivalent

typedef float v2f __attribute__((ext_vector_type(2)));
typedef float v8f __attribute__((ext_vector_type(8)));

// BCE on probability (cls path: p is already a probability; no divide)
__device__ __forceinline__ float bce_prob(float p, float y) {
    p = fminf(fmaxf(p, EPSF), 1.0f - EPSF);
    return -(y * __logf(p) + (1.0f - y) * __logf(1.0f - p));
}

// BCE(sigmoid(x), y) without any division:
//   -log(sigmoid(x))   = softplus(-x)
//   -log(1-sigmoid(x)) = softplus(x)
//   softplus(t) = max(t,0) + log1p(exp(-|t|))
// clamp(p,eps,1-eps) == cap each term at -log(eps).
__device__ __forceinline__ float bce_logit(float x, float y) {
    const float sp  = __logf(1.0f + __expf(-fabsf(x)));  // shared log1p(e^-|x|)
    const float spp = fminf(fmaxf(x, 0.0f) + sp, BCE_CAP);   // softplus(x)
    const float spn = fminf(fmaxf(-x, 0.0f) + sp, BCE_CAP);  // softplus(-x)
    return y * spn + (1.0f - y) * spp;
}

// ---------------------------------------------------------------------------
// ws layout (floats): [0..7] mask-BCE sums (atomic), [8..15] cls-pos sums,
// [16..23] cls-neg sums, [24..31] loc sums.
// ---------------------------------------------------------------------------
__global__ void zero_ws_kernel(float* __restrict__ ws) {
    if (threadIdx.x < NN) ws[threadIdx.x] = 0.0f;
}

// ---------------------------------------------------------------------------
// Classification + localization losses: one block per image, gather-reduce.
// ---------------------------------------------------------------------------
__global__ __launch_bounds__(256) void cls_loc_kernel(
    const float* __restrict__ map_class,     // [N,A]
    const float* __restrict__ map_box,       // [N,A,4]
    const float* __restrict__ anchor_center, // [A,2]
    const float* __restrict__ anchor_hw,     // [A,2]
    const float* __restrict__ gt_boxes,      // [N,B,4]
    const int*   __restrict__ pos_idx,       // [N,K]
    const int*   __restrict__ neg_idx,       // [N,3K]
    const int*   __restrict__ gt_idx,        // [N,K]
    float* __restrict__ ws) {
    const int n   = blockIdx.x;
    const int tid = threadIdx.x;

    float acc_pos = 0.0f, acc_neg = 0.0f, acc_loc = 0.0f;
    const float L10E = 0.43429448190325176f;  // log10(e)

    for (int k = tid; k < KK; k += blockDim.x) {
        const int pa = pos_idx[n * KK + k];
        // positive BCE vs 1
        acc_pos += bce_prob(map_class[n * AA + pa], 1.0f);
        // localization: SmoothL1(pred - encoded target)
        const float* pr = map_box + ((size_t)n * AA + pa) * 4;
        const float  ac0 = anchor_center[pa * 2 + 0];
        const float  ac1 = anchor_center[pa * 2 + 1];
        const float  ah  = anchor_hw[pa * 2 + 0];
        const float  aw  = anchor_hw[pa * 2 + 1];
        const int    gb  = gt_idx[n * KK + k];
        const float* g   = gt_boxes + ((size_t)n * BB + gb) * 4;
        float t[4];
        t[0] = (g[0] - ac0) / ah;
        t[1] = (g[1] - ac1) / aw;
        t[2] = __logf(g[2] / ah) * L10E;
        t[3] = __logf(g[3] / aw) * L10E;
#pragma unroll
        for (int c = 0; c < 4; ++c) {
            const float d  = pr[c] - t[c];
            const float ad = fabsf(d);
            acc_loc += (ad < 1.0f) ? 0.5f * d * d : ad - 0.5f;
        }
    }
    for (int j = tid; j < 3 * KK; j += blockDim.x) {
        const int na = neg_idx[n * 3 * KK + j];
        acc_neg += bce_prob(map_class[n * AA + na], 0.0f);
    }

    __shared__ float sA[256], sB[256], sC[256];
    sA[tid] = acc_pos; sB[tid] = acc_neg; sC[tid] = acc_loc;
    __syncthreads();
    for (int s = 128; s > 0; s >>= 1) {
        if (tid < s) {
            sA[tid] += sA[tid + s];
            sB[tid] += sB[tid + s];
            sC[tid] += sC[tid + s];
        }
        __syncthreads();
    }
    if (tid == 0) {
        ws[8 + n]  = sA[0];
        ws[16 + n] = sB[0];
        ws[24 + n] = sC[0];
    }
}

// ---------------------------------------------------------------------------
// Mask loss, fused: WMMA f32 16x16x4 GEMM tile (coef[16x4] x proto[4x16])
// -> logit-space BCE (no divide) vs gathered gt mask -> wave reduce -> atomic.
// Never materializes the 105 MB pred tensor.
// grid = (MTIL, NN); block = 256 threads = 8 wave32s.
// ---------------------------------------------------------------------------
__global__ __launch_bounds__(256) void mask_wmma_kernel(
    const float* __restrict__ map_coef, // [N,A,P]
    const float* __restrict__ proto,    // [N,P,HW,HW]
    const float* __restrict__ gt_masks, // [N,B,HW,HW]
    const int*   __restrict__ pos_idx,  // [N,K]
    const int*   __restrict__ gt_idx,   // [N,K]
    float* __restrict__ ws) {
    const int m_tile = blockIdx.x;   // 0..12
    const int n      = blockIdx.y;   // 0..7
    const int tid    = threadIdx.x;
    const int lane   = tid & 31;
    const int wave   = tid >> 5;
    const int lcol   = lane & 15;    // M for A-frag, N-col for B/C frags
    const int khalf  = lane >> 4;    // selects K={0,1} vs K={2,3} rows

    // LDS: gt-mask row base for the 16 anchor rows of this tile, plus validity
    __shared__ int s_gtbase[16];
    if (tid < 16) {
        const int k = m_tile * 16 + tid;
        s_gtbase[tid] = (k < KK)
            ? (int)(((size_t)n * BB + gt_idx[n * KK + k]) * PIX)
            : -1;
    }
    __syncthreads();

    // A fragment: ISA layout — lane holds row M=lane%16; VGPR0=K(0|2), VGPR1=K(1|3)
    v2f a; a.x = 0.0f; a.y = 0.0f;
    {
        const int k = m_tile * 16 + lcol;
        if (k < KK) {
            const int pa = pos_idx[n * KK + k];
            const float* cf = map_coef + ((size_t)n * AA + pa) * PP + 2 * khalf;
            a.x = cf[0];
            a.y = cf[1];
        }
    }

    // Branchless per-row epilogue params: safe base + 0/1 weight
    int   rbase[8];
    float rw[8];
#pragma unroll
    for (int v = 0; v < 8; ++v) {
        const int info = s_gtbase[v + 8 * khalf];
        rw[v]    = (info >= 0) ? 1.0f : 0.0f;
        rbase[v] = (info >= 0) ? info : 0;
    }

    // B rows of proto this lane-half reads (K = 2*khalf, 2*khalf+1)
    const float* prow0 = proto + ((size_t)n * PP + 2 * khalf) * PIX;
    const float* prow1 = prow0 + PIX;

    float acc = 0.0f;
    for (int t = wave; t < NTIL; t += 8) {
        const int col = t * 16 + lcol;
        // prefetch next tile for this wave
        __builtin_prefetch(prow0 + col + 8 * 16, 0, 1);
        v2f b; b.x = prow0[col]; b.y = prow1[col];

        v8f c = {};
        // D = A(16x4) x B(4x16) + 0 : full P-reduction in one instruction
        c = __builtin_amdgcn_wmma_f32_16x16x4_f32(
                /*neg_a=*/false, a, /*neg_b=*/false, b,
                /*c_mod=*/(short)0, c, /*reuse_a=*/false, /*reuse_b=*/false);

        // Straight-line epilogue: 8 coalesced gathers, then divide-free BCE
        float yv[8];
#pragma unroll
        for (int v = 0; v < 8; ++v)
            yv[v] = gt_masks[rbase[v] + col];
#pragma unroll
        for (int v = 0; v < 8; ++v)
            acc += rw[v] * bce_logit(c[v], yv[v]);
    }

    // wave32 reduction, then one atomic per wave
    for (int off = 16; off > 0; off >>= 1)
        acc += __shfl_down(acc, off, 32);
    if (lane == 0)
        atomicAdd(&ws[n], acc);
}

// ---------------------------------------------------------------------------
// Combine: per_img = (cls_pos/K + cls_neg/3K)/K + loc/K + (msk_sum/PIX)/K,
// then mean over batch.
// ---------------------------------------------------------------------------
__global__ void final_kernel(const float* __restrict__ ws, float* __restrict__ out) {
    if (threadIdx.x == 0 && blockIdx.x == 0) {
        float s = 0.0f;
        const float invK = 1.0f / (float)KK;
        for (int n = 0; n < NN; ++n) {
            const float cls = ws[8 + n] * invK + ws[16 + n] * (1.0f / (3.0f * KK));
            const float loc = ws[24 + n];
            const float msk = ws[n] * (1.0f / (float)PIX);
            s += (cls + loc + msk) * invK;
        }
        out[0] = s / (float)NN;
    }
}

extern "C" void kernel_launch(void* const* d_in, const int* in_sizes, int n_in,
                              void* d_out, int out_size, void* d_ws, size_t ws_size,
                              hipStream_t stream) {
    const float* map_class     = (const float*)d_in[0];
    const float* map_box       = (const float*)d_in[1];
    const float* map_coef      = (const float*)d_in[2];
    const float* proto         = (const float*)d_in[3];
    const float* anchor_center = (const float*)d_in[4];
    const float* anchor_hw     = (const float*)d_in[5];
    const float* gt_boxes      = (const float*)d_in[6];
    const float* gt_masks      = (const float*)d_in[7];
    const int*   pos_idx       = (const int*)d_in[8];
    const int*   neg_idx       = (const int*)d_in[9];
    const int*   gt_idx        = (const int*)d_in[10];
    float*       ws            = (float*)d_ws;
    float*       out           = (float*)d_out;

    zero_ws_kernel<<<1, 32, 0, stream>>>(ws);
    cls_loc_kernel<<<NN, 256, 0, stream>>>(map_class, map_box, anchor_center,
                                           anchor_hw, gt_boxes, pos_idx, neg_idx,
                                           gt_idx, ws);
    mask_wmma_kernel<<<dim3(MTIL, NN), 256, 0, stream>>>(map_coef, proto, gt_masks,
                                                         pos_idx, gt_idx, ws);
    final_kernel<<<1, 32, 0, stream>>>(ws, out);
}